// ImprovedVectorQuantizer_7773890806040
// MI455X (gfx1250) — compile-verified
//
#include <hip/hip_runtime.h>
#include <hip/hip_bf16.h>

// VQ quantizer for MI455X (gfx1250, wave32, WMMA).
// inputs: [32, 64, 16384] f32, emb: [128, 64] f32
// out: quantized [32,64,16384] f32 (33,554,432) ++ indices [524288] (as float)
//
// Memory-bound: ~270MB traffic -> ~11.6us floor at 23.3 TB/s. GEMM part
// (8.6 GFLOP) done with V_WMMA_F32_16X16X4_F32; codebook lives in LDS.

#define T_DIM 16384
#define D_DIM 64
#define K_CODES 128
#define QUANT_ELEMS (32ull * D_DIM * T_DIM)   // 33,554,432

typedef float v2f __attribute__((ext_vector_type(2)));
typedef float v8f __attribute__((ext_vector_type(8)));

__global__ __launch_bounds__(256) void vq_wmma_kernel(
    const float* __restrict__ x,     // [32, 64, 16384]
    const float* __restrict__ emb,   // [128, 64]
    float* __restrict__ out)         // quantized ++ indices
{
    __shared__ float Elds[D_DIM * K_CODES];   // codebook transposed: E[d][k], 32KB
    __shared__ float enorm[K_CODES];          // ||e_k||^2
    __shared__ int   idxs[8 * 16];            // per-wave winning code per sample

    const int tid  = threadIdx.x;
    const int lane = tid & 31;
    const int wave = tid >> 5;
    const int m    = lane & 15;   // sample-in-tile (rows M) / column N / t offset
    const int half = lane >> 4;   // K-pair selector in WMMA fragments

    // ---- cooperative codebook load (global [k][d] -> LDS [d][k]) ----
    for (int i = tid; i < D_DIM * K_CODES; i += 256) {
        int k = i >> 6;            // 0..127
        int d = i & (D_DIM - 1);   // 0..63
        Elds[d * K_CODES + k] = emb[i];
    }
    __syncthreads();
    if (tid < K_CODES) {
        float s = 0.f;
        for (int d = 0; d < D_DIM; ++d) {
            float e = Elds[d * K_CODES + tid];
            s += e * e;
        }
        enorm[tid] = s;
    }
    __syncthreads();

    // ---- tile assignment: one wave = 16 samples (16 consecutive t, one b) ----
    const int tile = blockIdx.x * 8 + wave;      // 0..32767
    const int n0   = tile * 16;                  // flat sample base, n = b*T + t
    const int b    = n0 >> 14;                   // n0 / 16384
    const int t0   = n0 & (T_DIM - 1);
    const float* xb = x + (size_t)b * D_DIM * T_DIM + t0;

    // ---- load A fragments once: 16 K-slices of 4 dims, 2 VGPRs each ----
    // A layout (16x4 f32): lanes 0-15 rows M; VGPR0 = K0 / K2 (halves), VGPR1 = K1 / K3
    float a0[16], a1[16];
#pragma unroll
    for (int s = 0; s < 16; ++s) {
        int d0 = 4 * s + 2 * half;
        a0[s] = __builtin_nontemporal_load(&xb[(size_t)d0 * T_DIM + m]);
        a1[s] = __builtin_nontemporal_load(&xb[(size_t)(d0 + 1) * T_DIM + m]);
    }

    // ---- GEMM over 8 code-tiles, folding into running argmin ----
    float bv[8];
    int   bi[8];
#pragma unroll
    for (int r = 0; r < 8; ++r) { bv[r] = 3.402823466e+38f; bi[r] = 0; }

    for (int kt = 0; kt < 8; ++kt) {
        v8f acc = {0.f, 0.f, 0.f, 0.f, 0.f, 0.f, 0.f, 0.f};
        float ek = enorm[kt * 16 + m];   // this lane's column's ||e||^2
#pragma unroll
        for (int s = 0; s < 16; ++s) {
            int d0 = 4 * s + 2 * half;
            v2f a = {a0[s], a1[s]};
            // B layout (4x16 f32) mirrors A: lanes 0-15 cols N; VGPR0=K0/K2, VGPR1=K1/K3
            v2f bb = { Elds[d0 * K_CODES + kt * 16 + m],
                       Elds[(d0 + 1) * K_CODES + kt * 16 + m] };
            acc = __builtin_amdgcn_wmma_f32_16x16x4_f32(
                false, a, false, bb, (short)0, acc, false, false);
        }
        // C/D layout: VGPR r holds row M = r + 8*half, col N = m
#pragma unroll
        for (int r = 0; r < 8; ++r) {
            float score = ek - 2.0f * acc[r];
            int code = kt * 16 + m;
            if (score < bv[r]) { bv[r] = score; bi[r] = code; }  // strict < : first-min wins over kt
        }
    }

    // ---- argmin reduction across the 16 columns (lanes within each half) ----
#pragma unroll
    for (int r = 0; r < 8; ++r) {
#pragma unroll
        for (int mask = 8; mask >= 1; mask >>= 1) {
            float ov = __shfl_xor(bv[r], mask, 32);
            int   oi = __shfl_xor(bi[r], mask, 32);
            if (ov < bv[r] || (ov == bv[r] && oi < bi[r])) { bv[r] = ov; bi[r] = oi; }
        }
    }

    // ---- publish indices (sample M = r + 8*half) ----
    float* outIdx = out + QUANT_ELEMS;
#pragma unroll
    for (int r = 0; r < 8; ++r) {
        if (m == r) {
            int sample = r + 8 * half;            // 0..15
            idxs[wave * 16 + sample] = bi[r];
            outIdx[n0 + sample] = (float)bi[r];   // indices emitted as float values
        }
    }

    // ---- gather winning codewords from LDS and stream out ----
    // quantized[b, d, t0+m] = E[idx[m]][d]; lane m handles sample m, halves split d parity
    int myidx = idxs[wave * 16 + m];
    float* outq = out + (size_t)b * D_DIM * T_DIM + t0;
#pragma unroll
    for (int dd = 0; dd < 32; ++dd) {
        int d = 2 * dd + half;
        float v = Elds[d * K_CODES + myidx];
        __builtin_nontemporal_store(v, &outq[(size_t)d * T_DIM + m]);
    }
}

extern "C" void kernel_launch(void* const* d_in, const int* in_sizes, int n_in,
                              void* d_out, int out_size, void* d_ws, size_t ws_size,
                              hipStream_t stream) {
    const float* x   = (const float*)d_in[0];   // inputs [32,64,16384]
    const float* emb = (const float*)d_in[1];   // emb_weight [128,64]
    float* out = (float*)d_out;

    // 32768 sample-tiles of 16; 8 waves (256 threads) per block -> 4096 blocks
    vq_wmma_kernel<<<dim3(4096), dim3(256), 0, stream>>>(x, emb, out);
}